// SimpleMoleculeGCN_10299331575891
// MI455X (gfx1250) — compile-verified
//
#include <hip/hip_runtime.h>
#include <hip/hip_bf16.h>
#include <math.h>

#define EPSF 1e-5f

typedef __attribute__((ext_vector_type(2))) float v2f;
typedef __attribute__((ext_vector_type(8))) float v8f;

// ---------------------------------------------------------------------------
// Generic fp32 WMMA GEMM: C[M x Nc] = A[M x K] @ B[K x Nc] (+ bias[Nc])
// One wave computes one 16x16 output tile; K stepped by 4 via
// V_WMMA_F32_16X16X4_F32. Requires M, Nc multiples of 16 and K multiple of 4
// (true for all call sites here: 100000x64x64, 512x192x128, 512x128x64).
// EXEC stays all-ones inside the wave (guards are wave-uniform).
// ---------------------------------------------------------------------------
__global__ void gemm_f32_wmma(const float* __restrict__ A,
                              const float* __restrict__ B,
                              const float* __restrict__ bias,
                              float* __restrict__ C,
                              int M, int K, int Nc) {
  int wave = blockIdx.x * (blockDim.x >> 5) + (threadIdx.x >> 5);
  int tilesN = Nc >> 4;
  int totalTiles = (M >> 4) * tilesN;
  if (wave >= totalTiles) return;           // wave-uniform
  int lane = threadIdx.x & 31;
  int l16  = lane & 15;
  int hi   = lane >> 4;                     // 0: lanes 0-15, 1: lanes 16-31
  int tileM = wave / tilesN;
  int tileN = wave - tileM * tilesN;
  int row0 = tileM << 4;
  int col0 = tileN << 4;

  const float* arow = A + (size_t)(row0 + l16) * K;
  const float* bcol = B + (size_t)col0 + l16;

  v8f c = {};
  #pragma unroll 4
  for (int k0 = 0; k0 < K; k0 += 4) {
    // A 16x4 layout: VGPR0 = {K=k0 (lanes0-15), K=k0+2 (lanes16-31)}, VGPR1 = {k0+1, k0+3}
    v2f a, b;
    a.x = arow[k0 + 2 * hi];
    a.y = arow[k0 + 2 * hi + 1];
    // B 4x16 layout (row striped across lanes): VGPR0 = {K=k0, K=k0+2}, VGPR1 = {k0+1, k0+3}
    b.x = bcol[(size_t)(k0 + 2 * hi) * Nc];
    b.y = bcol[(size_t)(k0 + 2 * hi + 1) * Nc];
    c = __builtin_amdgcn_wmma_f32_16x16x4_f32(false, a, false, b,
                                              (short)0, c, false, false);
  }

  float bv = bias ? bias[col0 + l16] : 0.0f;
  // C 16x16 f32 layout: VGPR j -> row (j + 8*hi), col = l16
  #pragma unroll
  for (int j = 0; j < 8; ++j) {
    int r = row0 + j + hi * 8;
    C[(size_t)r * Nc + col0 + l16] = c[j] + bv;
  }
}

// ---------------------------------------------------------------------------
// Degree / normalization
// ---------------------------------------------------------------------------
__global__ void deg_init(float* deg, int N) {
  int i = blockIdx.x * blockDim.x + threadIdx.x;
  if (i < N) deg[i] = 1.0f;                 // self loop
}
__global__ void deg_edges(const int* __restrict__ dst, float* deg, int E) {
  int e = blockIdx.x * blockDim.x + threadIdx.x;
  if (e < E) atomicAdd(&deg[dst[e]], 1.0f);
}
__global__ void deg_rsqrt(float* deg, int N) {
  int i = blockIdx.x * blockDim.x + threadIdx.x;
  if (i < N) deg[i] = rsqrtf(deg[i]);       // deg >= 1 always
}

// ---------------------------------------------------------------------------
// agg[n] = convB + dinv[n]^2 * hw[n]   (self-loop term + bias init)
// 16 threads per node, float4 per thread (D = 64)
// ---------------------------------------------------------------------------
__global__ void agg_init(const float* __restrict__ hw, const float* __restrict__ dinv,
                         const float* __restrict__ convB, float* __restrict__ agg, int N) {
  int t = blockIdx.x * blockDim.x + threadIdx.x;
  int n = t >> 4;
  if (n >= N) return;
  int c4 = (t & 15) * 4;
  float di = dinv[n];
  float w = di * di;
  float4 hv = *(const float4*)(hw + (size_t)n * 64 + c4);
  float4 bb = *(const float4*)(convB + c4);
  float4 r;
  r.x = bb.x + w * hv.x; r.y = bb.y + w * hv.y;
  r.z = bb.z + w * hv.z; r.w = bb.w + w * hv.w;
  *(float4*)(agg + (size_t)n * 64 + c4) = r;
}

// ---------------------------------------------------------------------------
// Edge scatter: agg[dst] += dinv[src]*dinv[dst] * hw[src]
// 16 threads per edge, float4 gather + 4 float atomics.
// ---------------------------------------------------------------------------
__global__ void edge_scatter(const int* __restrict__ src, const int* __restrict__ dst,
                             const float* __restrict__ dinv, const float* __restrict__ hw,
                             float* __restrict__ agg, int E) {
  int t = blockIdx.x * blockDim.x + threadIdx.x;
  int e = t >> 4;
  if (e >= E) return;
  int c4 = (t & 15) * 4;
  int s = src[e], d = dst[e];
  float w = dinv[s] * dinv[d];
  float4 hv = *(const float4*)(hw + (size_t)s * 64 + c4);
  float* ap = agg + (size_t)d * 64 + c4;
  atomicAdd(ap + 0, w * hv.x);
  atomicAdd(ap + 1, w * hv.y);
  atomicAdd(ap + 2, w * hv.z);
  atomicAdd(ap + 3, w * hv.w);
}

// ---------------------------------------------------------------------------
// BatchNorm stats: per-column sum & sumsq over N rows -> sums[0:64], sums[64:128]
// ---------------------------------------------------------------------------
__global__ void bn_stats(const float* __restrict__ agg, float* __restrict__ sums, int N) {
  int col = threadIdx.x & 63;
  int rg  = threadIdx.x >> 6;               // 0..3
  float s = 0.f, ss = 0.f;
  for (int n = blockIdx.x * 4 + rg; n < N; n += gridDim.x * 4) {
    float v = agg[(size_t)n * 64 + col];
    s += v; ss += v * v;
  }
  __shared__ float ls[256], lss[256];
  ls[threadIdx.x] = s; lss[threadIdx.x] = ss;
  __syncthreads();
  if (rg == 0) {
    s  = ls[col]  + ls[col + 64]  + ls[col + 128]  + ls[col + 192];
    ss = lss[col] + lss[col + 64] + lss[col + 128] + lss[col + 192];
    atomicAdd(&sums[col], s);
    atomicAdd(&sums[64 + col], ss);
  }
}

__global__ void bn_finalize(const float* __restrict__ sums, float* __restrict__ stats,
                            float invN) {
  int c = threadIdx.x;                      // 64 threads
  float mu = sums[c] * invN;
  float var = sums[64 + c] * invN - mu * mu;
  stats[c] = mu;
  stats[64 + c] = rsqrtf(var + EPSF);
}

// ---------------------------------------------------------------------------
// Fused BN-apply + (residual) + LayerNorm + exact GELU. One wave per row,
// 2 columns per lane (D = 64), shfl_xor wave reductions (wave32).
// ---------------------------------------------------------------------------
__global__ void bn_res_ln_gelu(const float* __restrict__ agg, const float* __restrict__ stats,
                               const float* __restrict__ bng, const float* __restrict__ bnb,
                               const float* __restrict__ lng, const float* __restrict__ lnb,
                               const float* __restrict__ xin, float* __restrict__ out, int N) {
  int row = blockIdx.x * (blockDim.x >> 5) + (threadIdx.x >> 5);
  if (row >= N) return;
  int lane = threadIdx.x & 31;
  float v[2];
  float s = 0.f;
  #pragma unroll
  for (int j = 0; j < 2; ++j) {
    int c = lane + 32 * j;
    float a = agg[(size_t)row * 64 + c];
    float h = (a - stats[c]) * stats[64 + c] * bng[c] + bnb[c];
    if (xin) h += xin[(size_t)row * 64 + c];
    v[j] = h; s += h;
  }
  for (int m = 16; m >= 1; m >>= 1) s += __shfl_xor(s, m, 32);
  float mu = s * (1.0f / 64.0f);
  float q = 0.f;
  #pragma unroll
  for (int j = 0; j < 2; ++j) { v[j] -= mu; q += v[j] * v[j]; }
  for (int m = 16; m >= 1; m >>= 1) q += __shfl_xor(q, m, 32);
  float rs = rsqrtf(q * (1.0f / 64.0f) + EPSF);
  #pragma unroll
  for (int j = 0; j < 2; ++j) {
    int c = lane + 32 * j;
    float x = v[j] * rs * lng[c] + lnb[c];
    out[(size_t)row * 64 + c] = 0.5f * x * (1.0f + erff(x * 0.70710678118654752f));
  }
}

// ---------------------------------------------------------------------------
// Pooling: monotone uint encoding for float atomicMax
// ---------------------------------------------------------------------------
__device__ __forceinline__ unsigned fenc(float f) {
  unsigned u = __float_as_uint(f);
  return (u & 0x80000000u) ? ~u : (u | 0x80000000u);
}
__device__ __forceinline__ float fdec(unsigned u) {
  return (u & 0x80000000u) ? __uint_as_float(u & 0x7FFFFFFFu) : __uint_as_float(~u);
}

__global__ void pool_accum(const float* __restrict__ h, const int* __restrict__ batch,
                           float* __restrict__ s, unsigned* __restrict__ mxe,
                           float* __restrict__ cnt, int N) {
  int t = blockIdx.x * blockDim.x + threadIdx.x;
  int n = t >> 4;
  if (n >= N) return;
  int c4 = (t & 15) * 4;
  int g = batch[n];
  float4 hv = *(const float4*)(h + (size_t)n * 64 + c4);
  float* sp = s + (size_t)g * 64 + c4;
  unsigned* mp = mxe + (size_t)g * 64 + c4;
  atomicAdd(sp + 0, hv.x); atomicMax(mp + 0, fenc(hv.x));
  atomicAdd(sp + 1, hv.y); atomicMax(mp + 1, fenc(hv.y));
  atomicAdd(sp + 2, hv.z); atomicMax(mp + 2, fenc(hv.z));
  atomicAdd(sp + 3, hv.w); atomicMax(mp + 3, fenc(hv.w));
  if ((t & 15) == 0) atomicAdd(&cnt[g], 1.0f);
}

__global__ void pool_finish(const float* __restrict__ s, const unsigned* __restrict__ mxe,
                            const float* __restrict__ cnt, float* __restrict__ z, int G) {
  int t = blockIdx.x * blockDim.x + threadIdx.x;
  if (t >= G * 64) return;
  int g = t >> 6, c = t & 63;
  float cn = cnt[g];
  float sv = s[(size_t)g * 64 + c];
  float mean = sv / fmaxf(cn, 1.0f);
  float mx = (cn > 0.f) ? fdec(mxe[(size_t)g * 64 + c]) : 0.0f;
  float* zr = z + (size_t)g * 192;
  zr[c] = mean; zr[64 + c] = mx; zr[128 + c] = sv;
}

// ---------------------------------------------------------------------------
// Row LayerNorm + exact GELU (in-place capable). One wave per row; W in {64,128}.
// ---------------------------------------------------------------------------
__global__ void ln_gelu_rows(const float* __restrict__ in, const float* __restrict__ g,
                             const float* __restrict__ b, float* __restrict__ out,
                             int M, int W) {
  int row = blockIdx.x * (blockDim.x >> 5) + (threadIdx.x >> 5);
  if (row >= M) return;
  int lane = threadIdx.x & 31;
  int cpl = W >> 5;                         // 2 or 4
  float v[4];
  float s = 0.f;
  const float* rp = in + (size_t)row * W;
  for (int j = 0; j < cpl; ++j) { v[j] = rp[lane + j * 32]; s += v[j]; }
  for (int m = 16; m >= 1; m >>= 1) s += __shfl_xor(s, m, 32);
  float mu = s / (float)W;
  float q = 0.f;
  for (int j = 0; j < cpl; ++j) { v[j] -= mu; q += v[j] * v[j]; }
  for (int m = 16; m >= 1; m >>= 1) q += __shfl_xor(q, m, 32);
  float rs = rsqrtf(q / (float)W + EPSF);
  float* op = out + (size_t)row * W;
  for (int j = 0; j < cpl; ++j) {
    int c = lane + j * 32;
    float x = v[j] * rs * g[c] + b[c];
    op[c] = 0.5f * x * (1.0f + erff(x * 0.70710678118654752f));
  }
}

// ---------------------------------------------------------------------------
// Final projection: out[g] = dot(z2[g,:64], W3) + b3
// ---------------------------------------------------------------------------
__global__ void final_proj(const float* __restrict__ z2, const float* __restrict__ W3,
                           const float* __restrict__ b3, float* __restrict__ out, int G) {
  int row = blockIdx.x * (blockDim.x >> 5) + (threadIdx.x >> 5);
  if (row >= G) return;
  int lane = threadIdx.x & 31;
  const float* zr = z2 + (size_t)row * 64;
  float s = zr[lane] * W3[lane] + zr[lane + 32] * W3[lane + 32];
  for (int m = 16; m >= 1; m >>= 1) s += __shfl_xor(s, m, 32);
  if (lane == 0) out[row] = s + b3[0];
}

// ---------------------------------------------------------------------------
extern "C" void kernel_launch(void* const* d_in, const int* in_sizes, int n_in,
                              void* d_out, int out_size, void* d_ws, size_t ws_size,
                              hipStream_t stream) {
  (void)in_sizes; (void)n_in; (void)out_size; (void)ws_size;
  const int N = 100000, E = 1600000, D = 64, G = 512, L = 3;

  const float* x     = (const float*)d_in[0];
  const int*   ei    = (const int*)d_in[1];
  const int*   batch = (const int*)d_in[2];
  const float* convW = (const float*)d_in[3];
  const float* convB = (const float*)d_in[4];
  const float* bn_g  = (const float*)d_in[5];
  const float* bn_b  = (const float*)d_in[6];
  const float* ln_g  = (const float*)d_in[7];
  const float* ln_b  = (const float*)d_in[8];
  const float* W1    = (const float*)d_in[9];
  const float* b1    = (const float*)d_in[10];
  const float* ln1_g = (const float*)d_in[11];
  const float* ln1_b = (const float*)d_in[12];
  const float* W2    = (const float*)d_in[13];
  const float* b2    = (const float*)d_in[14];
  const float* ln2_g = (const float*)d_in[15];
  const float* ln2_b = (const float*)d_in[16];
  const float* W3    = (const float*)d_in[17];
  const float* b3    = (const float*)d_in[18];
  float* out = (float*)d_out;

  // Workspace layout (floats)
  float* wf = (float*)d_ws;
  size_t ND = (size_t)N * D;
  float*    bufA     = wf;
  float*    bufB     = wf + ND;
  float*    agg      = wf + 2 * ND;
  float*    dinv     = wf + 3 * ND;               // N
  float*    bnsum    = dinv + N;                  // 128
  float*    bnstat   = bnsum + 128;               // 128
  float*    pool_s   = bnstat + 128;              // G*64
  float*    pool_cnt = pool_s + (size_t)G * 64;   // G
  unsigned* pool_mx  = (unsigned*)(pool_cnt + G); // G*64
  float*    z        = (float*)(pool_mx + (size_t)G * 64); // G*192
  float*    z1       = z + (size_t)G * 192;       // G*128
  float*    z2       = z1 + (size_t)G * 128;      // G*64

  const int* src = ei;
  const int* dst = ei + E;

  // Degrees (with self loops) -> dinv = deg^{-1/2}
  deg_init <<<(N + 255) / 256, 256, 0, stream>>>(dinv, N);
  deg_edges<<<(E + 255) / 256, 256, 0, stream>>>(dst, dinv, E);
  deg_rsqrt<<<(N + 255) / 256, 256, 0, stream>>>(dinv, N);

  const float* hcur = x;
  float* hwb = bufA;      // GEMM output, later reused as h_next
  float* other = bufB;
  for (int i = 0; i < L; ++i) {
    int tiles = (N / 16) * (D / 16);  // 25000 wave-tiles
    gemm_f32_wmma<<<(tiles + 7) / 8, 256, 0, stream>>>(
        hcur, convW + (size_t)i * D * D, nullptr, hwb, N, D, D);
    agg_init<<<(N * 16 + 255) / 256, 256, 0, stream>>>(hwb, dinv, convB + i * D, agg, N);
    edge_scatter<<<(E * 16 + 255) / 256, 256, 0, stream>>>(src, dst, dinv, hwb, agg, E);
    hipMemsetAsync(bnsum, 0, 128 * sizeof(float), stream);
    bn_stats<<<256, 256, 0, stream>>>(agg, bnsum, N);
    bn_finalize<<<1, 64, 0, stream>>>(bnsum, bnstat, 1.0f / (float)N);
    bn_res_ln_gelu<<<(N + 7) / 8, 256, 0, stream>>>(
        agg, bnstat, bn_g + i * D, bn_b + i * D, ln_g + i * D, ln_b + i * D,
        (i > 0) ? hcur : nullptr, hwb, N);   // hw is dead; reuse as h_next
    hcur = hwb;
    float* t = hwb; hwb = other; other = t;  // ping-pong
  }

  // Pooling (s, cnt, mx are contiguous -> single memset; 0 == minimal fenc value)
  hipMemsetAsync(pool_s, 0, (size_t)(G * 64 + G + G * 64) * 4, stream);
  pool_accum<<<(N * 16 + 255) / 256, 256, 0, stream>>>(hcur, batch, pool_s, pool_mx, pool_cnt, N);
  pool_finish<<<(G * 64 + 255) / 256, 256, 0, stream>>>(pool_s, pool_mx, pool_cnt, z, G);

  // MLP head (all WMMA)
  int t1 = (G / 16) * (128 / 16);
  gemm_f32_wmma<<<(t1 + 7) / 8, 256, 0, stream>>>(z, W1, b1, z1, G, 3 * D, 2 * D);
  ln_gelu_rows<<<(G + 7) / 8, 256, 0, stream>>>(z1, ln1_g, ln1_b, z1, G, 2 * D);
  int t2 = (G / 16) * (64 / 16);
  gemm_f32_wmma<<<(t2 + 7) / 8, 256, 0, stream>>>(z1, W2, b2, z2, G, 2 * D, D);
  ln_gelu_rows<<<(G + 7) / 8, 256, 0, stream>>>(z2, ln2_g, ln2_b, z2, G, D);
  final_proj<<<(G + 7) / 8, 256, 0, stream>>>(z2, W3, b3, out, G);
}